// OnlineCosineTripletLoss_48833778155618
// MI455X (gfx1250) — compile-verified
//
#include <hip/hip_runtime.h>

// Guard only in the DEVICE pass: the host (x86) pass never has amdgcn builtins.
#if defined(__HIP_DEVICE_COMPILE__)
#if !__has_builtin(__builtin_amdgcn_wmma_f32_16x16x4_f32)
#error "gfx1250 device pass: __builtin_amdgcn_wmma_f32_16x16x4_f32 not available"
#endif
#endif

typedef __attribute__((ext_vector_type(2))) float v2f;
typedef __attribute__((ext_vector_type(8))) float v8f;

#define BDIM 512
#define DDIM 512
#define MARGIN_F 0.1f

// ---------------------------------------------------------------------------
// Kernel 0: zero the accumulators (deterministic; harness does not re-poison)
// ---------------------------------------------------------------------------
__global__ void init_accum_kernel(float* __restrict__ accum_sum,
                                  unsigned int* __restrict__ accum_cnt) {
    if (threadIdx.x == 0) {
        accum_sum[0] = 0.0f;
        accum_cnt[0] = 0u;
    }
}

// ---------------------------------------------------------------------------
// Kernel 1: per-row inverse L2 norms for A (blocks 0..511) and B (512..1023).
// 256 threads/block (8 waves of 32); each thread loads one float2 of the row.
// ---------------------------------------------------------------------------
__global__ void inv_norm_kernel(const float* __restrict__ A,
                                const float* __restrict__ B,
                                float* __restrict__ invA,
                                float* __restrict__ invB) {
    const int row   = blockIdx.x;                 // 0..1023
    const bool isA  = (row < BDIM);
    const int r     = isA ? row : row - BDIM;
    const float* src = isA ? A : B;

    const v2f* p = (const v2f*)(src + (size_t)r * DDIM);
    v2f x = p[threadIdx.x];                       // 256 * 2 = 512 floats
    float ss = x.x * x.x + x.y * x.y;

    // wave32 butterfly reduction
    #pragma unroll
    for (int off = 16; off > 0; off >>= 1)
        ss += __shfl_xor(ss, off, 32);

    __shared__ float red[8];
    const int warp = threadIdx.x >> 5;
    if ((threadIdx.x & 31) == 0) red[warp] = ss;
    __syncthreads();

    if (threadIdx.x == 0) {
        float tot = 0.0f;
        #pragma unroll
        for (int w = 0; w < 8; ++w) tot += red[w];
        const float inv = 1.0f / sqrtf(tot);
        if (isA) invA[r] = inv; else invB[r] = inv;
    }
}

// ---------------------------------------------------------------------------
// Kernel 2: sim = diag(invA) * (A @ B^T) * diag(invB), exact f32 via
// V_WMMA_F32_16X16X4_F32. One wave per 16x16 tile; K-loop 512 in steps of 4.
//
// f32 16x4 A-frag layout (ISA 7.12.2): lane l owns M = (l&15); half-wave 0
// holds K={k0,k0+1}, half-wave 1 holds K={k0+2,k0+3} -> one 8B load per iter.
// B-frag (4x16, element (k,n) = Bmat[n][k]) has the identical addressing with
// N = (l&15). C/D: VGPR v in lane l = D[M = v + 8*(l>>4)][N = (l&15)].
// ---------------------------------------------------------------------------
__global__ void __launch_bounds__(32)
wmma_sim_kernel(const float* __restrict__ A,
                const float* __restrict__ B,
                const float* __restrict__ invA,
                const float* __restrict__ invB,
                float* __restrict__ S) {
    const int lane  = threadIdx.x;            // 0..31, EXEC all ones (no divergence)
    const int mt    = blockIdx.x << 4;
    const int nt    = blockIdx.y << 4;
    const int r     = lane & 15;
    const int koff  = (lane >> 4) << 1;       // 0 or 2

    const float* arow = A + (size_t)(mt + r) * DDIM + koff;
    const float* brow = B + (size_t)(nt + r) * DDIM + koff;

    v8f c = {};
    #pragma unroll 8
    for (int k0 = 0; k0 < DDIM; k0 += 4) {
        v2f av = *(const v2f*)(arow + k0);
        v2f bv = *(const v2f*)(brow + k0);
        // 8 args: (neg_a, A, neg_b, B, c_mod, C, reuse_a, reuse_b)
        c = __builtin_amdgcn_wmma_f32_16x16x4_f32(
                false, av, false, bv, (short)0, c, false, false);
    }

    // Epilogue: apply the two row/col inverse norms (fused normalization).
    const float sb   = invB[nt + r];          // this lane's column scale
    const int   mrow = mt + ((lane >> 4) << 3);
    float* out = S + (size_t)mrow * BDIM + nt + r;
    #pragma unroll
    for (int v = 0; v < 8; ++v) {
        const float scale = invA[mrow + v] * sb;   // broadcast load per half-wave
        out[(size_t)v * BDIM] = c[v] * scale;
    }
}

// ---------------------------------------------------------------------------
// Kernel 3: per anchor row a:  sum/count of relu(sim[a,n] - sim[a,a] + m), n!=a
// ---------------------------------------------------------------------------
__global__ void triplet_reduce_kernel(const float* __restrict__ S,
                                      float* __restrict__ accum_sum,
                                      unsigned int* __restrict__ accum_cnt) {
    const int a = blockIdx.x;
    const float* row = S + (size_t)a * BDIM;
    const float diag = row[a];

    float lsum = 0.0f;
    unsigned int lcnt = 0u;
    for (int n = threadIdx.x; n < BDIM; n += 256) {
        const float t = row[n] - diag + MARGIN_F;
        if ((n != a) && (t > 0.0f)) { lsum += t; lcnt++; }
    }

    #pragma unroll
    for (int off = 16; off > 0; off >>= 1) {
        lsum += __shfl_xor(lsum, off, 32);
        lcnt += __shfl_xor(lcnt, off, 32);
    }

    __shared__ float        ssum[8];
    __shared__ unsigned int scnt[8];
    const int warp = threadIdx.x >> 5;
    if ((threadIdx.x & 31) == 0) { ssum[warp] = lsum; scnt[warp] = lcnt; }
    __syncthreads();

    if (threadIdx.x == 0) {
        float ts = 0.0f; unsigned int tc = 0u;
        #pragma unroll
        for (int w = 0; w < 8; ++w) { ts += ssum[w]; tc += scnt[w]; }
        atomicAdd(accum_sum, ts);
        atomicAdd(accum_cnt, tc);
    }
}

// ---------------------------------------------------------------------------
// Kernel 4: final scalar
// ---------------------------------------------------------------------------
__global__ void finalize_kernel(const float* __restrict__ accum_sum,
                                const unsigned int* __restrict__ accum_cnt,
                                float* __restrict__ out) {
    if (threadIdx.x == 0) {
        const unsigned int c = accum_cnt[0];
        const float denom = (c > 0u) ? (float)c : (float)(BDIM * (BDIM - 1));
        out[0] = accum_sum[0] / denom;
    }
}

// ---------------------------------------------------------------------------
extern "C" void kernel_launch(void* const* d_in, const int* in_sizes, int n_in,
                              void* d_out, int out_size, void* d_ws, size_t ws_size,
                              hipStream_t stream) {
    (void)in_sizes; (void)n_in; (void)out_size; (void)ws_size;

    const float* A = (const float*)d_in[0];   // in_embeddings  [512,512] f32
    const float* B = (const float*)d_in[1];   // out_embeddings [512,512] f32

    char* ws = (char*)d_ws;
    float*        S         = (float*)ws;                                   // 1 MB sim
    float*        invA      = (float*)(ws + (size_t)BDIM * DDIM * sizeof(float));
    float*        invB      = invA + BDIM;
    float*        accum_sum = invB + BDIM;
    unsigned int* accum_cnt = (unsigned int*)(accum_sum + 1);

    init_accum_kernel<<<1, 32, 0, stream>>>(accum_sum, accum_cnt);
    inv_norm_kernel<<<2 * BDIM, 256, 0, stream>>>(A, B, invA, invB);

    dim3 gemm_grid(BDIM / 16, BDIM / 16);     // 32x32 tiles, one wave each
    wmma_sim_kernel<<<gemm_grid, 32, 0, stream>>>(A, B, invA, invB, S);

    triplet_reduce_kernel<<<BDIM, 256, 0, stream>>>(S, accum_sum, accum_cnt);
    finalize_kernel<<<1, 32, 0, stream>>>(accum_sum, accum_cnt, (float*)d_out);
}